// hn_DGCNN_decoder_hier2_7232724926917
// MI455X (gfx1250) — compile-verified
//
#include <hip/hip_runtime.h>

// ---------------- constants ----------------
#define LRELU_NS 0.2f
#define EPSF 1e-6f
constexpr int B = 4, NQ = 20000, KNN = 20, CF = 16;
constexpr int TOT = B * NQ;   // 80000

typedef __attribute__((ext_vector_type(2))) float v2f;
typedef __attribute__((ext_vector_type(8))) float v8f;

__device__ __forceinline__ float leaky(float x) { return x >= 0.f ? x : LRELU_NS * x; }

// Vector-Neuron leaky ReLU on one 3-vector (whole vector kept, or its part along d scaled by LRELU_NS).
__device__ __forceinline__ void vn_lrelu3(float& p0, float& p1, float& p2,
                                          float d0, float d1, float d2) {
  float dot = p0 * d0 + p1 * d1 + p2 * d2;
  float dsq = d0 * d0 + d1 * d1 + d2 * d2 + EPSF;
  if (dot < 0.f) {
    float f = (1.f - LRELU_NS) * dot / dsq;
    p0 -= f * d0; p1 -= f * d1; p2 -= f * d2;
  }
}

__device__ __forceinline__ v8f wmma4(v2f a, v2f b, v8f c) {
  // D = A(16x4,f32) * B(4x16,f32) + C(16x16,f32)
  return __builtin_amdgcn_wmma_f32_16x16x4_f32(false, a, false, b, (short)0, c,
                                               false, false);
}

// ---------------- kNN ----------------
template<int N, int BPB>
__global__ void __launch_bounds__(128) knn_kernel(const float* __restrict__ p,
                                                  const float* __restrict__ pp,
                                                  int* __restrict__ idx) {
  __shared__ float sx[N], sy[N], sz[N];
  int b = blockIdx.x / BPB;
  int qb = (blockIdx.x % BPB) * 128;
  const float* ppb = pp + (size_t)b * 3 * N;
  for (int i = threadIdx.x; i < N; i += 128) {
    sx[i] = ppb[i]; sy[i] = ppb[N + i]; sz[i] = ppb[2 * N + i];
  }
  __syncthreads();
  int q = qb + threadIdx.x;
  if (q >= NQ) return;
  const float* pq = p + ((size_t)b * NQ + q) * 3;
  float qx = pq[0], qy = pq[1], qz = pq[2];
  float bd[KNN]; int bi[KNN];
#pragma unroll
  for (int t = 0; t < KNN; ++t) { bd[t] = 3.0e38f; bi[t] = 0; }
  for (int r = 0; r < N; ++r) {
    float dx = qx - sx[r], dy = qy - sy[r], dz = qz - sz[r];
    float d2 = dx * dx + dy * dy + dz * dz;
    if (d2 < bd[KNN - 1]) {
      float cd = d2; int ci = r;
#pragma unroll
      for (int t = 0; t < KNN; ++t) {
        if (cd < bd[t]) {
          float td = bd[t]; int ti = bi[t];
          bd[t] = cd; bi[t] = ci; cd = td; ci = ti;
        }
      }
    }
  }
  int* ob = idx + ((size_t)b * NQ + q) * KNN;
#pragma unroll
  for (int t = 0; t < KNN; ++t) ob[t] = bi[t];
}

// ---------------- fused level: graph-feature + conv*_1/2/3 + pool ----------------
template<int N, bool HASPREV>
__global__ void __launch_bounds__(256) level_kernel(
    const float* __restrict__ p, const float* __restrict__ pp,
    const float* __restrict__ fea, const float* __restrict__ sfea,
    const int* __restrict__ idx,
    const float* __restrict__ xprev, const float* __restrict__ sprev,
    const float* __restrict__ Wv1, const float* __restrict__ Wd1,
    const float* __restrict__ Wv2, const float* __restrict__ Wd2,
    const float* __restrict__ Ws2,
    const float* __restrict__ Wv3, const float* __restrict__ Wd3,
    const float* __restrict__ Ws3,
    float* __restrict__ xout, float* __restrict__ sout) {
  constexpr int CV2 = HASPREV ? 24 : 20;     // geo(4)+fea(16)[+xprev(4)]
  constexpr int CS2 = HASPREV ? 68 : 36;     // syf(32)[+sprev(32)]+norms(4)
  constexpr int NOFF = CS2 - 4;
  __shared__ float sWv1[8], sWd1[8];
  __shared__ float sWv2[4 * CV2], sWd2[4 * CV2], sWs2[32 * CS2];
  __shared__ float sWv3[16], sWd3[16], sWs3[32 * 36];
  for (int i = threadIdx.x; i < 8; i += 256) { sWv1[i] = Wv1[i]; sWd1[i] = Wd1[i]; }
  for (int i = threadIdx.x; i < 4 * CV2; i += 256) { sWv2[i] = Wv2[i]; sWd2[i] = Wd2[i]; }
  for (int i = threadIdx.x; i < 32 * CS2; i += 256) sWs2[i] = Ws2[i];
  for (int i = threadIdx.x; i < 16; i += 256) { sWv3[i] = Wv3[i]; sWd3[i] = Wd3[i]; }
  for (int i = threadIdx.x; i < 32 * 36; i += 256) sWs3[i] = Ws3[i];
  __syncthreads();
  int g = blockIdx.x * 256 + threadIdx.x;
  if (g >= TOT) return;
  int b = g / NQ, q = g % NQ;
  const float* pq = p + (size_t)g * 3;
  float x0 = pq[0], x1 = pq[1], x2 = pq[2];

  // k-invariant contributions from xprev / sprev
  float avp[12], adp[12], sbias[32];
#pragma unroll
  for (int i = 0; i < 12; ++i) { avp[i] = 0.f; adp[i] = 0.f; }
#pragma unroll
  for (int i = 0; i < 32; ++i) sbias[i] = 0.f;
  if constexpr (HASPREV) {
#pragma unroll
    for (int c = 0; c < 4; ++c) {
      float f0 = xprev[((size_t)(b * 4 + c) * 3 + 0) * NQ + q];
      float f1 = xprev[((size_t)(b * 4 + c) * 3 + 1) * NQ + q];
      float f2 = xprev[((size_t)(b * 4 + c) * 3 + 2) * NQ + q];
#pragma unroll
      for (int o = 0; o < 4; ++o) {
        float wv = sWv2[o * CV2 + 20 + c], wd = sWd2[o * CV2 + 20 + c];
        avp[o * 3] += wv * f0; avp[o * 3 + 1] += wv * f1; avp[o * 3 + 2] += wv * f2;
        adp[o * 3] += wd * f0; adp[o * 3 + 1] += wd * f1; adp[o * 3 + 2] += wd * f2;
      }
    }
    for (int c = 0; c < 32; ++c) {
      float sc = sprev[((size_t)b * 32 + c) * NQ + q];
#pragma unroll
      for (int o = 0; o < 32; ++o) sbias[o] += sWs2[o * CS2 + 32 + c] * sc;
    }
  }

  float vm[12];
#pragma unroll
  for (int i = 0; i < 12; ++i) vm[i] = 0.f;
  float sm[32];
#pragma unroll
  for (int i = 0; i < 32; ++i) sm[i] = -3.0e38f;

  const int* ib = idx + (size_t)g * KNN;
  const float* ppb = pp + (size_t)b * 3 * N;
  const float* feab = fea + (size_t)b * CF * 3 * N;
  const float* sfb = sfea + (size_t)b * 32 * N;

  for (int k = 0; k < KNN; ++k) {
    int j = ib[k];
    float y0 = ppb[j], y1 = ppb[N + j], y2 = ppb[2 * N + j];
    float e0 = y0 - x0, e1 = y1 - x1, e2 = y2 - x2;
    // conv*_1 : vn_linear_lrelu([y-x, x])  (2 -> 4 channels)
    float geo[12];
#pragma unroll
    for (int o = 0; o < 4; ++o) {
      float wv0 = sWv1[o * 2], wv1 = sWv1[o * 2 + 1];
      float wd0 = sWd1[o * 2], wd1 = sWd1[o * 2 + 1];
      float p0 = wv0 * e0 + wv1 * x0, p1 = wv0 * e1 + wv1 * x1, p2 = wv0 * e2 + wv1 * x2;
      float d0 = wd0 * e0 + wd1 * x0, d1 = wd0 * e1 + wd1 * x1, d2 = wd0 * e2 + wd1 * x2;
      vn_lrelu3(p0, p1, p2, d0, d1, d2);
      geo[o * 3] = p0; geo[o * 3 + 1] = p1; geo[o * 3 + 2] = p2;
    }
    // conv*_2 vector branch
    float av[12], ad[12];
#pragma unroll
    for (int i = 0; i < 12; ++i) { av[i] = avp[i]; ad[i] = adp[i]; }
#pragma unroll
    for (int c = 0; c < 4; ++c) {
      float f0 = geo[c * 3], f1 = geo[c * 3 + 1], f2 = geo[c * 3 + 2];
#pragma unroll
      for (int o = 0; o < 4; ++o) {
        float wv = sWv2[o * CV2 + c], wd = sWd2[o * CV2 + c];
        av[o * 3] += wv * f0; av[o * 3 + 1] += wv * f1; av[o * 3 + 2] += wv * f2;
        ad[o * 3] += wd * f0; ad[o * 3 + 1] += wd * f1; ad[o * 3 + 2] += wd * f2;
      }
    }
    for (int c = 0; c < CF; ++c) {
      float f0 = feab[(c * 3 + 0) * N + j];
      float f1 = feab[(c * 3 + 1) * N + j];
      float f2 = feab[(c * 3 + 2) * N + j];
#pragma unroll
      for (int o = 0; o < 4; ++o) {
        float wv = sWv2[o * CV2 + 4 + c], wd = sWd2[o * CV2 + 4 + c];
        av[o * 3] += wv * f0; av[o * 3 + 1] += wv * f1; av[o * 3 + 2] += wv * f2;
        ad[o * 3] += wd * f0; ad[o * 3 + 1] += wd * f1; ad[o * 3 + 2] += wd * f2;
      }
    }
    float v2c[12], nr2[4];
#pragma unroll
    for (int o = 0; o < 4; ++o) {
      float p0 = av[o * 3], p1 = av[o * 3 + 1], p2 = av[o * 3 + 2];
      vn_lrelu3(p0, p1, p2, ad[o * 3], ad[o * 3 + 1], ad[o * 3 + 2]);
      v2c[o * 3] = p0; v2c[o * 3 + 1] = p1; v2c[o * 3 + 2] = p2;
      nr2[o] = sqrtf(p0 * p0 + p1 * p1 + p2 * p2 + EPSF);
    }
    // conv*_2 scalar branch
    float as[32];
#pragma unroll
    for (int o = 0; o < 32; ++o) as[o] = sbias[o];
    for (int c = 0; c < 32; ++c) {
      float sc = sfb[c * N + j];
#pragma unroll
      for (int o = 0; o < 32; ++o) as[o] += sWs2[o * CS2 + c] * sc;
    }
#pragma unroll
    for (int c = 0; c < 4; ++c) {
      float sc = nr2[c];
#pragma unroll
      for (int o = 0; o < 32; ++o) as[o] += sWs2[o * CS2 + NOFF + c] * sc;
    }
#pragma unroll
    for (int o = 0; o < 32; ++o) as[o] = leaky(as[o]);
    // conv*_3 vector branch (4 -> 4)
    float v3[12], nr3[4];
#pragma unroll
    for (int o = 0; o < 4; ++o) {
      float p0 = 0.f, p1 = 0.f, p2 = 0.f, d0 = 0.f, d1 = 0.f, d2 = 0.f;
#pragma unroll
      for (int c = 0; c < 4; ++c) {
        float wv = sWv3[o * 4 + c], wd = sWd3[o * 4 + c];
        p0 += wv * v2c[c * 3]; p1 += wv * v2c[c * 3 + 1]; p2 += wv * v2c[c * 3 + 2];
        d0 += wd * v2c[c * 3]; d1 += wd * v2c[c * 3 + 1]; d2 += wd * v2c[c * 3 + 2];
      }
      vn_lrelu3(p0, p1, p2, d0, d1, d2);
      v3[o * 3] = p0; v3[o * 3 + 1] = p1; v3[o * 3 + 2] = p2;
      nr3[o] = sqrtf(p0 * p0 + p1 * p1 + p2 * p2 + EPSF);
    }
    // conv*_3 scalar branch (36 -> 32)
    float s3[32];
#pragma unroll
    for (int o = 0; o < 32; ++o) s3[o] = 0.f;
    for (int c = 0; c < 32; ++c) {
      float sc = as[c];
#pragma unroll
      for (int o = 0; o < 32; ++o) s3[o] += sWs3[o * 36 + c] * sc;
    }
#pragma unroll
    for (int c = 0; c < 4; ++c) {
      float sc = nr3[c];
#pragma unroll
      for (int o = 0; o < 32; ++o) s3[o] += sWs3[o * 36 + 32 + c] * sc;
    }
#pragma unroll
    for (int i = 0; i < 12; ++i) vm[i] += v3[i];
#pragma unroll
    for (int o = 0; o < 32; ++o) sm[o] = fmaxf(sm[o], leaky(s3[o]));
  }
  const float inv_k = 1.f / (float)KNN;
#pragma unroll
  for (int o = 0; o < 4; ++o)
#pragma unroll
    for (int d = 0; d < 3; ++d)
      xout[((size_t)(b * 4 + o) * 3 + d) * NQ + q] = vm[o * 3 + d] * inv_k;
#pragma unroll
  for (int o = 0; o < 32; ++o) sout[((size_t)b * 32 + o) * NQ + q] = sm[o];
}

// ---------------- conv1x1 + vn_std(z1) -> c[TOT][48] ----------------
__global__ void __launch_bounds__(256) conv1x1_kernel(
    const float* __restrict__ x1, const float* __restrict__ s1,
    const float* __restrict__ x2, const float* __restrict__ s2,
    const float* __restrict__ x3, const float* __restrict__ s3,
    const float* __restrict__ Wv, const float* __restrict__ Wd,
    const float* __restrict__ Ws, const float* __restrict__ z1w,
    float* __restrict__ cbuf) {
  __shared__ float sWv[16 * 12], sWd[16 * 12], sWs[32 * 112], sz[16];
  for (int i = threadIdx.x; i < 192; i += 256) { sWv[i] = Wv[i]; sWd[i] = Wd[i]; }
  for (int i = threadIdx.x; i < 32 * 112; i += 256) sWs[i] = Ws[i];
  for (int i = threadIdx.x; i < 16; i += 256) sz[i] = z1w[i];
  __syncthreads();
  int g = blockIdx.x * 256 + threadIdx.x;
  if (g >= TOT) return;
  int b = g / NQ, q = g % NQ;
  float vin[36];
#pragma unroll
  for (int c = 0; c < 4; ++c)
#pragma unroll
    for (int d = 0; d < 3; ++d) {
      vin[c * 3 + d]       = x1[((size_t)(b * 4 + c) * 3 + d) * NQ + q];
      vin[(4 + c) * 3 + d] = x2[((size_t)(b * 4 + c) * 3 + d) * NQ + q];
      vin[(8 + c) * 3 + d] = x3[((size_t)(b * 4 + c) * 3 + d) * NQ + q];
    }
  float cv[48], nr[16];
#pragma unroll
  for (int o = 0; o < 16; ++o) {
    float p0 = 0.f, p1 = 0.f, p2 = 0.f, d0 = 0.f, d1 = 0.f, d2 = 0.f;
#pragma unroll
    for (int c = 0; c < 12; ++c) {
      float wv = sWv[o * 12 + c], wd = sWd[o * 12 + c];
      p0 += wv * vin[c * 3]; p1 += wv * vin[c * 3 + 1]; p2 += wv * vin[c * 3 + 2];
      d0 += wd * vin[c * 3]; d1 += wd * vin[c * 3 + 1]; d2 += wd * vin[c * 3 + 2];
    }
    vn_lrelu3(p0, p1, p2, d0, d1, d2);
    cv[o * 3] = p0; cv[o * 3 + 1] = p1; cv[o * 3 + 2] = p2;
    nr[o] = sqrtf(p0 * p0 + p1 * p1 + p2 * p2 + EPSF);
  }
  float acc[32];
#pragma unroll
  for (int o = 0; o < 32; ++o) {
    float a = 0.f;
#pragma unroll
    for (int c = 0; c < 16; ++c) a += sWs[o * 112 + 96 + c] * nr[c];
    acc[o] = a;
  }
  for (int c = 0; c < 32; ++c) {
    float a1 = s1[((size_t)b * 32 + c) * NQ + q];
    float a2 = s2[((size_t)b * 32 + c) * NQ + q];
    float a3 = s3[((size_t)b * 32 + c) * NQ + q];
#pragma unroll
    for (int o = 0; o < 32; ++o)
      acc[o] += sWs[o * 112 + c] * a1 + sWs[o * 112 + 32 + c] * a2 +
                sWs[o * 112 + 64 + c] * a3;
  }
  // vn_std with z1
  float zx = 0.f, zy = 0.f, zz = 0.f;
#pragma unroll
  for (int c = 0; c < 16; ++c) {
    zx += sz[c] * cv[c * 3]; zy += sz[c] * cv[c * 3 + 1]; zz += sz[c] * cv[c * 3 + 2];
  }
  float zl = sqrtf(zx * zx + zy * zy + zz * zz);
  float scl = 1.f / (zl + EPSF);
  zx *= scl; zy *= scl; zz *= scl;
  float* co = cbuf + (size_t)g * 48;
#pragma unroll
  for (int c = 0; c < 16; ++c)
    co[c] = cv[c * 3] * zx + cv[c * 3 + 1] * zy + cv[c * 3 + 2] * zz;
#pragma unroll
  for (int o = 0; o < 32; ++o) co[16 + o] = leaky(acc[o]);
}

// ---------------- fc_p1_1 + fc_p1_2 + vn_std(z2) + fc_p2 -> net[TOT][32] ----------------
__global__ void __launch_bounds__(256) fcp_kernel(
    const float* __restrict__ p,
    const float* Wv11, const float* Wd11, const float* Ws11,
    const float* Wv12, const float* Wd12, const float* Ws12,
    const float* z2w, const float* Wp2, const float* bp2,
    float* __restrict__ netbuf) {
  __shared__ float sWv11[16], sWd11[16], sWs11[24 * 16];
  __shared__ float sWv12[256], sWd12[256], sWs12[24 * 40];
  __shared__ float sz2[16], sWp2[32 * 40], sbp2[32];
  for (int i = threadIdx.x; i < 16; i += 256) { sWv11[i] = Wv11[i]; sWd11[i] = Wd11[i]; sz2[i] = z2w[i]; }
  for (int i = threadIdx.x; i < 24 * 16; i += 256) sWs11[i] = Ws11[i];
  for (int i = threadIdx.x; i < 256; i += 256) { sWv12[i] = Wv12[i]; sWd12[i] = Wd12[i]; }
  for (int i = threadIdx.x; i < 24 * 40; i += 256) sWs12[i] = Ws12[i];
  for (int i = threadIdx.x; i < 32 * 40; i += 256) sWp2[i] = Wp2[i];
  for (int i = threadIdx.x; i < 32; i += 256) sbp2[i] = bp2[i];
  __syncthreads();
  int g = blockIdx.x * 256 + threadIdx.x;
  if (g >= TOT) return;
  float q0 = p[(size_t)g * 3], q1 = p[(size_t)g * 3 + 1], q2 = p[(size_t)g * 3 + 2];
  float nv1[48], nr1[16];
#pragma unroll
  for (int o = 0; o < 16; ++o) {
    float p0 = sWv11[o] * q0, p1 = sWv11[o] * q1, p2 = sWv11[o] * q2;
    float d0 = sWd11[o] * q0, d1 = sWd11[o] * q1, d2 = sWd11[o] * q2;
    vn_lrelu3(p0, p1, p2, d0, d1, d2);
    nv1[o * 3] = p0; nv1[o * 3 + 1] = p1; nv1[o * 3 + 2] = p2;
    nr1[o] = sqrtf(p0 * p0 + p1 * p1 + p2 * p2 + EPSF);
  }
  float ns1[24];
#pragma unroll
  for (int o = 0; o < 24; ++o) {
    float a = 0.f;
#pragma unroll
    for (int c = 0; c < 16; ++c) a += sWs11[o * 16 + c] * nr1[c];
    ns1[o] = leaky(a);
  }
  float nv2[48], nr2[16];
#pragma unroll
  for (int o = 0; o < 16; ++o) {
    float p0 = 0.f, p1 = 0.f, p2 = 0.f, d0 = 0.f, d1 = 0.f, d2 = 0.f;
#pragma unroll
    for (int c = 0; c < 16; ++c) {
      float wv = sWv12[o * 16 + c], wd = sWd12[o * 16 + c];
      p0 += wv * nv1[c * 3]; p1 += wv * nv1[c * 3 + 1]; p2 += wv * nv1[c * 3 + 2];
      d0 += wd * nv1[c * 3]; d1 += wd * nv1[c * 3 + 1]; d2 += wd * nv1[c * 3 + 2];
    }
    vn_lrelu3(p0, p1, p2, d0, d1, d2);
    nv2[o * 3] = p0; nv2[o * 3 + 1] = p1; nv2[o * 3 + 2] = p2;
    nr2[o] = sqrtf(p0 * p0 + p1 * p1 + p2 * p2 + EPSF);
  }
  float ns2[24];
#pragma unroll
  for (int o = 0; o < 24; ++o) {
    float a = 0.f;
#pragma unroll
    for (int c = 0; c < 24; ++c) a += sWs12[o * 40 + c] * ns1[c];
#pragma unroll
    for (int c = 0; c < 16; ++c) a += sWs12[o * 40 + 24 + c] * nr2[c];
    ns2[o] = leaky(a);
  }
  float zx = 0.f, zy = 0.f, zz = 0.f;
#pragma unroll
  for (int c = 0; c < 16; ++c) {
    zx += sz2[c] * nv2[c * 3]; zy += sz2[c] * nv2[c * 3 + 1]; zz += sz2[c] * nv2[c * 3 + 2];
  }
  float zl = sqrtf(zx * zx + zy * zy + zz * zz);
  float scl = 1.f / (zl + EPSF);
  zx *= scl; zy *= scl; zz *= scl;
  float inv[16];
#pragma unroll
  for (int c = 0; c < 16; ++c)
    inv[c] = nv2[c * 3] * zx + nv2[c * 3 + 1] * zy + nv2[c * 3 + 2] * zz;
  float* ob = netbuf + (size_t)g * 32;
#pragma unroll
  for (int o = 0; o < 32; ++o) {
    float a = sbp2[o];
#pragma unroll
    for (int c = 0; c < 16; ++c) a += sWp2[o * 40 + c] * inv[c];
#pragma unroll
    for (int c = 0; c < 24; ++c) a += sWp2[o * 40 + 16 + c] * ns2[c];
    ob[o] = a;
  }
}

// ---------------- decoder tail: 4x(fc_c + ResnetBlockFC) + fc_out, WMMA f32 ----------------
struct TailParams {
  const float* Wc[4]; const float* bc[4];
  const float* W0[4]; const float* b0[4];
  const float* W1[4]; const float* b1[4];
  const float* Wo;    const float* bo;
};

__global__ void __launch_bounds__(256) tail_kernel(const float* __restrict__ cbuf,
                                                   const float* __restrict__ netbuf,
                                                   TailParams P,
                                                   float* __restrict__ out) {
  // per wave: net(16x32) + c(16x48) + h(16x32) = 1792 floats
  __shared__ float lds[8 * 1792];
  const int wave = threadIdx.x >> 5;
  const int lane = threadIdx.x & 31;
  float* netT = lds + wave * 1792;
  float* cT = netT + 512;
  float* hT = cT + 768;
  const int m0 = (blockIdx.x * 8 + wave) * 16;   // TOT = 80000 = 16*5000 exactly
  for (int i = lane; i < 512; i += 32) netT[i] = netbuf[(size_t)m0 * 32 + i];
  for (int i = lane; i < 768; i += 32) cT[i] = cbuf[(size_t)m0 * 48 + i];
  __syncthreads();
  // A frag (16x4 f32): lanes 0-15 rows M=lane K={0,1}; lanes 16-31 rows M=lane-16 K={2,3}
  const int arow = lane & 15;
  const int koff = (lane >> 4) << 1;
  // C/D frag: vgpr r -> M = r (+8 for lanes 16-31), N = lane&15 (+16 per n-tile)
  const int mbase = (lane >> 4) << 3;
  const int ncl = lane & 15;
#pragma unroll
  for (int it = 0; it < 4; ++it) {
    const float* Wc = P.Wc[it]; const float* bc = P.bc[it];
    const float* W0 = P.W0[it]; const float* b0 = P.b0[it];
    const float* W1 = P.W1[it]; const float* b1 = P.b1[it];
    // net += c @ Wc^T + bc   (16x48 @ 48x32)
#pragma unroll
    for (int nt = 0; nt < 2; ++nt) {
      int ncol = ncl + nt * 16;
      v8f acc;
#pragma unroll
      for (int r = 0; r < 8; ++r) acc[r] = netT[(mbase + r) * 32 + ncol];
#pragma unroll
      for (int kc = 0; kc < 12; ++kc) {
        int kb = kc * 4 + koff;
        v2f a; a.x = cT[arow * 48 + kb]; a.y = cT[arow * 48 + kb + 1];
        v2f bb; bb.x = Wc[ncol * 48 + kb]; bb.y = Wc[ncol * 48 + kb + 1];
        acc = wmma4(a, bb, acc);
      }
      float bias = bc[ncol];
#pragma unroll
      for (int r = 0; r < 8; ++r) netT[(mbase + r) * 32 + ncol] = acc[r] + bias;
    }
    // h = relu(net) @ W0^T + b0
#pragma unroll
    for (int nt = 0; nt < 2; ++nt) {
      int ncol = ncl + nt * 16;
      float bias = b0[ncol];
      v8f acc;
#pragma unroll
      for (int r = 0; r < 8; ++r) acc[r] = bias;
#pragma unroll
      for (int kc = 0; kc < 8; ++kc) {
        int kb = kc * 4 + koff;
        v2f a; a.x = fmaxf(netT[arow * 32 + kb], 0.f);
        a.y = fmaxf(netT[arow * 32 + kb + 1], 0.f);
        v2f bb; bb.x = W0[ncol * 32 + kb]; bb.y = W0[ncol * 32 + kb + 1];
        acc = wmma4(a, bb, acc);
      }
#pragma unroll
      for (int r = 0; r < 8; ++r) hT[(mbase + r) * 32 + ncol] = acc[r];
    }
    // net += relu(h) @ W1^T + b1
#pragma unroll
    for (int nt = 0; nt < 2; ++nt) {
      int ncol = ncl + nt * 16;
      float bias = b1[ncol];
      v8f acc;
#pragma unroll
      for (int r = 0; r < 8; ++r) acc[r] = netT[(mbase + r) * 32 + ncol] + bias;
#pragma unroll
      for (int kc = 0; kc < 8; ++kc) {
        int kb = kc * 4 + koff;
        v2f a; a.x = fmaxf(hT[arow * 32 + kb], 0.f);
        a.y = fmaxf(hT[arow * 32 + kb + 1], 0.f);
        v2f bb; bb.x = W1[ncol * 32 + kb]; bb.y = W1[ncol * 32 + kb + 1];
        acc = wmma4(a, bb, acc);
      }
#pragma unroll
      for (int r = 0; r < 8; ++r) netT[(mbase + r) * 32 + ncol] = acc[r];
    }
  }
  // fc_out: leaky(net) @ Wo^T + bo
  if (lane < 16) {
    float acc = P.bo[0];
#pragma unroll
    for (int c = 0; c < 32; ++c) acc += leaky(netT[lane * 32 + c]) * P.Wo[c];
    out[m0 + lane] = acc;
  }
}

// ---------------- host launch ----------------
extern "C" void kernel_launch(void* const* d_in, const int* in_sizes, int n_in,
                              void* d_out, int out_size, void* d_ws, size_t ws_size,
                              hipStream_t stream) {
  (void)in_sizes; (void)out_size; (void)ws_size;
  const float* p     = (const float*)d_in[0];
  const float* pp    = (const float*)d_in[1];
  const float* fea   = (const float*)d_in[2];
  const float* sfea  = (const float*)d_in[3];
  const float* pp2   = (const float*)d_in[4];
  const float* fea2  = (const float*)d_in[5];
  const float* sfea2 = (const float*)d_in[6];
  const float* pp3   = (const float*)d_in[7];
  const float* fea3  = (const float*)d_in[8];
  const float* sfea3 = (const float*)d_in[9];
  const float* w[71] = {};
  for (int i = 0; i < 71 && 10 + i < n_in; ++i) w[i] = (const float*)d_in[10 + i];
  // params flattened in make_params insertion order
  const float *c11_Wv = w[0],  *c11_Wd = w[1];
  const float *c12_Wv = w[2],  *c12_Wd = w[3],  *c12_Ws = w[4];
  const float *c13_Wv = w[5],  *c13_Wd = w[6],  *c13_Ws = w[7];
  const float *c21_Wv = w[8],  *c21_Wd = w[9];
  const float *c22_Wv = w[10], *c22_Wd = w[11], *c22_Ws = w[12];
  const float *c23_Wv = w[13], *c23_Wd = w[14], *c23_Ws = w[15];
  const float *c31_Wv = w[16], *c31_Wd = w[17];
  const float *c32_Wv = w[18], *c32_Wd = w[19], *c32_Ws = w[20];
  const float *c33_Wv = w[21], *c33_Wd = w[22], *c33_Ws = w[23];
  const float *c41_Wv = w[24], *c41_Wd = w[25];
  const float *c42_Wv = w[26], *c42_Wd = w[27], *c42_Ws = w[28];
  const float *c43_Wv = w[29], *c43_Wd = w[30], *c43_Ws = w[31];
  const float *cx_Wv = w[32],  *cx_Wd = w[33],  *cx_Ws = w[34];
  const float *z1w = w[35];
  const float *p11_Wv = w[36], *p11_Wd = w[37], *p11_Ws = w[38];
  const float *p12_Wv = w[39], *p12_Wd = w[40], *p12_Ws = w[41];
  const float *z2w = w[42];
  const float *p2_W = w[43],  *p2_b = w[44];
  TailParams TP;
  for (int i = 0; i < 4; ++i) { TP.Wc[i] = w[45 + 2 * i]; TP.bc[i] = w[46 + 2 * i]; }
  for (int i = 0; i < 4; ++i) {
    TP.W0[i] = w[53 + 4 * i]; TP.b0[i] = w[54 + 4 * i];
    TP.W1[i] = w[55 + 4 * i]; TP.b1[i] = w[56 + 4 * i];
  }
  TP.Wo = w[69]; TP.bo = w[70];

  // workspace carving
  char* wsb = (char*)d_ws;
  auto carve = [&](size_t bytes) {
    void* r = (void*)wsb;
    wsb += (bytes + 255) & ~(size_t)255;
    return r;
  };
  int*   idxb = (int*)  carve((size_t)TOT * KNN * sizeof(int));
  float* x1b  = (float*)carve((size_t)B * 4 * 3 * NQ * sizeof(float));
  float* s1b  = (float*)carve((size_t)B * 32 * NQ * sizeof(float));
  float* x2b  = (float*)carve((size_t)B * 4 * 3 * NQ * sizeof(float));
  float* s2b  = (float*)carve((size_t)B * 32 * NQ * sizeof(float));
  float* x3ab = (float*)carve((size_t)B * 4 * 3 * NQ * sizeof(float));
  float* s3ab = (float*)carve((size_t)B * 32 * NQ * sizeof(float));
  float* x3bb = (float*)carve((size_t)B * 4 * 3 * NQ * sizeof(float));
  float* s3bb = (float*)carve((size_t)B * 32 * NQ * sizeof(float));
  float* cb   = (float*)carve((size_t)TOT * 48 * sizeof(float));
  float* nb   = (float*)carve((size_t)TOT * 32 * sizeof(float));

  constexpr int BPB = (NQ + 127) / 128;
  const dim3 knng(B * BPB), knnb(128);
  const int pblocks = (TOT + 255) / 256;

  // level 1 (N=1024, no xprev)
  knn_kernel<1024, BPB><<<knng, knnb, 0, stream>>>(p, pp, idxb);
  level_kernel<1024, false><<<pblocks, 256, 0, stream>>>(
      p, pp, fea, sfea, idxb, nullptr, nullptr,
      c11_Wv, c11_Wd, c12_Wv, c12_Wd, c12_Ws, c13_Wv, c13_Wd, c13_Ws, x1b, s1b);
  // level 2 (N=256)
  knn_kernel<256, BPB><<<knng, knnb, 0, stream>>>(p, pp2, idxb);
  level_kernel<256, true><<<pblocks, 256, 0, stream>>>(
      p, pp2, fea2, sfea2, idxb, x1b, s1b,
      c21_Wv, c21_Wd, c22_Wv, c22_Wd, c22_Ws, c23_Wv, c23_Wd, c23_Ws, x2b, s2b);
  // levels 3 & 4 share the same kNN (both query pp3)
  knn_kernel<64, BPB><<<knng, knnb, 0, stream>>>(p, pp3, idxb);
  level_kernel<64, true><<<pblocks, 256, 0, stream>>>(
      p, pp3, fea3, sfea3, idxb, x2b, s2b,
      c31_Wv, c31_Wd, c32_Wv, c32_Wd, c32_Ws, c33_Wv, c33_Wd, c33_Ws, x3ab, s3ab);
  level_kernel<64, true><<<pblocks, 256, 0, stream>>>(
      p, pp3, fea3, sfea3, idxb, x3ab, s3ab,
      c41_Wv, c41_Wd, c42_Wv, c42_Wd, c42_Ws, c43_Wv, c43_Wd, c43_Ws, x3bb, s3bb);
  // conv1x1 + vn_std(z1)
  conv1x1_kernel<<<pblocks, 256, 0, stream>>>(x1b, s1b, x2b, s2b, x3bb, s3bb,
                                              cx_Wv, cx_Wd, cx_Ws, z1w, cb);
  // fc_p path + fc_p2
  fcp_kernel<<<pblocks, 256, 0, stream>>>(p, p11_Wv, p11_Wd, p11_Ws,
                                          p12_Wv, p12_Wd, p12_Ws, z2w, p2_W, p2_b, nb);
  // WMMA decoder tail (80000 / (16*8) = 625 blocks)
  tail_kernel<<<TOT / 128, 256, 0, stream>>>(cb, nb, TP, (float*)d_out);
}